// knrm_39917426049514
// MI455X (gfx1250) — compile-verified
//
#include <hip/hip_runtime.h>
#include <hip/hip_bf16.h>

typedef __attribute__((ext_vector_type(2))) float v2f;
typedef __attribute__((ext_vector_type(8))) float v8f;

// One block per batch element. 256 threads = 8 wave32.
// Phase 1: stage q[32][300] to LDS, reduce sum-of-squares for q and d rows (ds_add_f32).
// Phase 2: per-wave fp32 WMMA (16x16x4) tiles of simT = d_tile x q_tile over K=300,
//          fused Gaussian-kernel pooling into per-lane acc[21] registers using a
//          geometric recurrence (2 fmul + 1 fadd per kernel instead of an exp each).
// Phase 3: ds-atomic merge into pooling[32][21], log/mask/sum -> feat, score.
//
// Kernel constants: mu = [1.0, 0.95, 0.85, ..., -0.95], sigma = [0.001, 0.1 x20].
// For k=1..20 (sigma=0.1): E(mu) = -50(s-mu)^2 is quadratic in mu, so walking mu in
// 0.1 steps gives g_{next} = g * r with r *= exp(-1) each step. Chains start at the
// center (mu = +/-0.05, worst-case start value e^-55, always representable) and walk
// outward, so intermediates never underflow before the true values do.
__global__ __launch_bounds__(256)
void knrm_wmma_kernel(const float* __restrict__ q_embed,   // [B,32,300]
                      const float* __restrict__ d_embed,   // [B,512,300]
                      const int*   __restrict__ mask_q,    // [B,32]
                      const int*   __restrict__ mask_d,    // [B,512]
                      const float* __restrict__ W,         // [21,1]
                      const float* __restrict__ bias,      // [1]
                      float* __restrict__ out,             // [B] score ++ [B,21] feat
                      int B)
{
    constexpr int Q = 32, D = 512, E = 300, K = 21;
    constexpr float EM1 = 0.36787944117144233f;   // e^-1

    __shared__ float q_lds[Q * E];      // 38400 B
    __shared__ float nqinv[Q];          // sumsq -> 1/max(norm,eps)
    __shared__ float ndinv[D];
    __shared__ float mdl[D];
    __shared__ float mql[Q];
    __shared__ float pooling[Q * K];
    __shared__ float featb[K];

    const int tid = threadIdx.x;
    const int b   = blockIdx.x;

    // ---- zero accumulators ----
    if (tid < Q) nqinv[tid] = 0.0f;
    for (int i = tid; i < D; i += 256) ndinv[i] = 0.0f;
    for (int i = tid; i < Q * K; i += 256) pooling[i] = 0.0f;
    __syncthreads();

    // ---- stage q into LDS + q row sum-of-squares (8 threads per row) ----
    {
        const int r  = tid >> 3;   // 0..31
        const int l8 = tid & 7;
        const float* qp = q_embed + ((size_t)b * Q + r) * E;
        float ss = 0.0f;
        for (int e = l8; e < E; e += 8) {
            float v = qp[e];
            q_lds[r * E + e] = v;
            ss = fmaf(v, v, ss);
        }
        atomicAdd(&nqinv[r], ss);
    }
    // ---- d row sum-of-squares (8 threads per row, 16 passes) ----
    {
        const int l8    = tid & 7;
        const int rbase = tid >> 3;  // 0..31
        for (int p = 0; p < 16; ++p) {
            const int r = rbase + 32 * p;
            const float* dp = d_embed + ((size_t)b * D + r) * E;
            float ss = 0.0f;
            for (int e = l8; e < E; e += 8) {
                float v = dp[e];
                ss = fmaf(v, v, ss);
            }
            atomicAdd(&ndinv[r], ss);
        }
    }
    // ---- masks to LDS ----
    for (int i = tid; i < D; i += 256) mdl[i] = (float)mask_d[(size_t)b * D + i];
    if (tid < Q) mql[tid] = (float)mask_q[(size_t)b * Q + tid];
    __syncthreads();

    // ---- sumsq -> reciprocal norms (matches x / max(||x||, 1e-12)) ----
    for (int i = tid; i < D; i += 256) {
        float n = fmaxf(sqrtf(ndinv[i]), 1e-12f);
        ndinv[i] = 1.0f / n;
    }
    if (tid < Q) {
        float n = fmaxf(sqrtf(nqinv[tid]), 1e-12f);
        nqinv[tid] = 1.0f / n;
    }
    __syncthreads();

    // ---- WMMA phase: simT tiles (M = d rows, N = q cols) ----
    const int wave = tid >> 5;        // 0..7
    const int lane = tid & 31;
    const int n16  = lane & 15;
    const int hi   = lane >> 4;       // 0 or 1
    const int q0   = (wave & 1) * 16;
    const int qcol = q0 + n16;        // this lane's q (N index)
    const float qinv = nqinv[qcol];

    float acc[K];
#pragma unroll
    for (int k = 0; k < K; ++k) acc[k] = 0.0f;

    // B-fragment source: B[k][n] = q[q0+n][k]; lanes 0-15 hold K pair {k0,k0+1},
    // lanes 16-31 hold {k0+2,k0+3} (mirrors the documented 16x4 fp32 A layout).
    const float* brow = q_lds + (size_t)qcol * E + 2 * hi;

    for (int dt = (wave >> 1); dt < D / 16; dt += 4) {
        const int d0 = dt * 16;
        // A-fragment: lane m and m+16 both hold d row (d0+m); K pairs {0,1}/{2,3}.
        const float* arow = d_embed + ((size_t)b * D + d0 + n16) * E + 2 * hi;

        v8f c = {0.f, 0.f, 0.f, 0.f, 0.f, 0.f, 0.f, 0.f};
#pragma unroll 5
        for (int k0 = 0; k0 < E; k0 += 4) {          // 75 WMMA steps
            v2f a  = *(const v2f*)(arow + k0);       // global_load_b64 (8B aligned)
            v2f bb = *(const v2f*)(brow + k0);       // ds_load_b64, bank-conflict-free
            c = __builtin_amdgcn_wmma_f32_16x16x4_f32(
                    false, a, false, bb, (short)0, c, false, false);
        }

        // fused Gaussian pooling: lane owns q = qcol, d rows d0 + i + 8*hi
#pragma unroll
        for (int i = 0; i < 8; ++i) {
            const int drow = d0 + i + 8 * hi;
            const float mdv = mdl[drow];
            if (mdv != 0.0f) {                        // skip masked-out docs
                const float s = c[i] * ndinv[drow] * qinv;

                // kernel 0: mu=1.0, sigma=0.001
                const float t0 = s - 1.0f;
                acc[0] = fmaf(__expf((t0 * -500000.0f) * t0), mdv, acc[0]);

                // chains start at mu = +/-0.05 (acc indices 10 / 11), walk outward
                const float du = s - 0.05f;
                float gu = mdv * __expf((du * -50.0f) * du);   // value at mu=+0.05
                const float dd = s + 0.05f;
                float gd = mdv * __expf((dd * -50.0f) * dd);   // value at mu=-0.05
                acc[10] += gu;
                acc[11] += gd;
                float ru = __expf(fmaf(10.0f, s, -1.0f));      // ratio toward mu=0.95
                float rd = __expf(fmaf(-10.0f, s, -1.0f));     // ratio toward mu=-0.95
#pragma unroll
                for (int j = 9; j >= 1; --j) {                 // mu = 0.15 .. 0.95
                    gu *= ru;
                    acc[j] += gu;
                    ru *= EM1;
                }
#pragma unroll
                for (int j = 12; j <= 20; ++j) {               // mu = -0.15 .. -0.95
                    gd *= rd;
                    acc[j] += gd;
                    rd *= EM1;
                }
            }
        }
    }

    // ---- merge per-lane kernels into pooling[q][k] ----
#pragma unroll
    for (int k = 0; k < K; ++k) atomicAdd(&pooling[qcol * K + k], acc[k]);
    __syncthreads();

    // ---- feat[k] = 0.01 * sum_q log(clip(pooling,1e-10)) * mask_q ----
    if (tid < K) {
        float f = 0.0f;
        for (int q = 0; q < Q; ++q) {
            float ps = fmaxf(pooling[q * K + tid], 1e-10f);
            f += __logf(ps) * mql[q];
        }
        f *= 0.01f;
        featb[tid] = f;
        out[B + (size_t)b * K + tid] = f;   // feat block after score block
    }
    __syncthreads();

    // ---- score[b] = feat . W + bias ----
    if (tid == 0) {
        float s = bias[0];
#pragma unroll
        for (int k = 0; k < K; ++k) s = fmaf(featb[k], W[k], s);
        out[b] = s;
    }
}

extern "C" void kernel_launch(void* const* d_in, const int* in_sizes, int n_in,
                              void* d_out, int out_size, void* d_ws, size_t ws_size,
                              hipStream_t stream) {
    const float* q_embed = (const float*)d_in[0];
    const float* d_embed = (const float*)d_in[1];
    const int*   mask_q  = (const int*)d_in[2];
    const int*   mask_d  = (const int*)d_in[3];
    const float* W       = (const float*)d_in[4];
    const float* bias    = (const float*)d_in[5];
    float*       out     = (float*)d_out;

    const int B = in_sizes[2] / 32;   // mask_q is [B,32]

    knrm_wmma_kernel<<<dim3(B), dim3(256), 0, stream>>>(
        q_embed, d_embed, mask_q, mask_d, W, bias, out, B);
}